// Decoder_59468117180498
// MI455X (gfx1250) — compile-verified
//
#include <hip/hip_runtime.h>
#include <hip/hip_bf16.h>
#include <math.h>

// Problem dims (compile-time constants from the reference)
#define TSTEPS 64
#define BATCH  128
#define SRC    512
#define EDIM   256
#define HDIM   512

// LDS staging geometry for the WMMA GEMMs
#define KC  32          // K-chunk per LDS stage
#define STR 36          // padded row stride (floats): 16B-aligned rows, bank-conflict-free

typedef float v2f __attribute__((ext_vector_type(2)));
typedef float v8f __attribute__((ext_vector_type(8)));

// D = A(16x4 f32) * B(4x16 f32) + C, full fp32 WMMA (CDNA5 V_WMMA_F32_16X16X4_F32)
__device__ __forceinline__ v8f wmma4_f32(v2f a, v2f b, v8f c) {
    return __builtin_amdgcn_wmma_f32_16x16x4_f32(
        /*neg_a=*/false, a, /*neg_b=*/false, b,
        /*c_mod=*/(short)0, c, /*reuse_a=*/false, /*reuse_b=*/false);
}

__device__ __forceinline__ float sigmoidf_(float x) {
    return 1.0f / (1.0f + __expf(-x));
}

// Low 32 bits of a flat shared-aperture address == LDS byte offset
__device__ __forceinline__ unsigned lds_off(const void* p) {
    return (unsigned)(uintptr_t)p;
}

// CDNA5 async DMA: global -> LDS, 16 bytes per lane, tracked by ASYNCcnt.
__device__ __forceinline__ void async_ld128(unsigned lds_byte, const float* gptr) {
    asm volatile("global_load_async_to_lds_b128 %0, %1, off"
                 :: "v"(lds_byte), "v"(gptr) : "memory");
}

__device__ __forceinline__ void wait_async_and_barrier() {
    asm volatile("s_wait_asynccnt 0x0" ::: "memory");
    __syncthreads();
}

// ---------------------------------------------------------------------------
// Block-cooperative GEMM panel with double-buffered async LDS staging.
//   Block = 256 threads (8 waves). Block output: 128 rows x 16 cols.
//   Wave w owns rows [16w, 16w+16); all waves share the same 16 columns, so the
//   B (weight) tile is DMA'd to LDS once per block. NB = number of 16-col
//   weight slices sharing the same A (1 for plain GEMM, 3 for the GRU gates).
//   acc[g] accumulates D for slice g across calls (lets callers split K).
// ---------------------------------------------------------------------------
template <int NB>
__device__ __forceinline__ void block_gemm_lds(
    const float* __restrict__ A, int lda, int ktot,        // A: 128 x ktot
    const float* __restrict__ B0, const float* __restrict__ B1,
    const float* __restrict__ B2, int ldb,                 // slices: 16 x ktot each
    v8f* acc,
    float* __restrict__ sA,                                // 2 * 128 * STR floats
    float* __restrict__ sB)                                // 2 * NB*16 * STR floats
{
    const int tid  = threadIdx.x;
    const int lane = tid & 31;
    const int wave = tid >> 5;
    const int l    = lane & 15;
    const int half = lane >> 4;
    const int ASZ  = 128 * STR;
    const int BSZ  = NB * 16 * STR;
    const float* Bb[3] = { B0, B1, B2 };
    const int nchunks = ktot / KC;

    // ---- prefetch chunk 0 ----
    {
        const unsigned aB = lds_off(sA);
        for (int i = tid; i < 128 * (KC / 4); i += 256) {
            const int row = i >> 3, seg = i & 7;
            async_ld128(aB + (unsigned)(row * STR + seg * 4) * 4u,
                        A + (size_t)row * lda + seg * 4);
        }
        const unsigned bB = lds_off(sB);
        for (int i = tid; i < NB * 16 * (KC / 4); i += 256) {
            const int row = i >> 3, seg = i & 7;
            async_ld128(bB + (unsigned)(row * STR + seg * 4) * 4u,
                        Bb[row >> 4] + (size_t)(row & 15) * ldb + seg * 4);
        }
    }
    wait_async_and_barrier();

    for (int c = 0; c < nchunks; ++c) {
        const int cur = c & 1, nxt = cur ^ 1;
        // ---- overlap: DMA chunk c+1 into the other buffer ----
        if (c + 1 < nchunks) {
            const int k0 = (c + 1) * KC;
            const unsigned aB = lds_off(sA + nxt * ASZ);
            for (int i = tid; i < 128 * (KC / 4); i += 256) {
                const int row = i >> 3, seg = i & 7;
                async_ld128(aB + (unsigned)(row * STR + seg * 4) * 4u,
                            A + (size_t)row * lda + k0 + seg * 4);
            }
            const unsigned bB = lds_off(sB + nxt * BSZ);
            for (int i = tid; i < NB * 16 * (KC / 4); i += 256) {
                const int row = i >> 3, seg = i & 7;
                async_ld128(bB + (unsigned)(row * STR + seg * 4) * 4u,
                            Bb[row >> 4] + (size_t)(row & 15) * ldb + k0 + seg * 4);
            }
        }
        // ---- WMMA stream from LDS (ds_load_b64 fragments, no loadcnt stalls) ----
        const float* aS = sA + cur * ASZ + (wave * 16 + l) * STR + 2 * half;
        const float* bS = sB + cur * BSZ + l * STR + 2 * half;
        #pragma unroll
        for (int k = 0; k < KC; k += 4) {
            const v2f a = *(const v2f*)(aS + k);
            #pragma unroll
            for (int g = 0; g < NB; ++g) {
                const v2f b = *(const v2f*)(bS + g * 16 * STR + k);
                acc[g] = wmma4_f32(a, b, acc[g]);
            }
        }
        wait_async_and_barrier();
    }
}

// ---------------------------------------------------------------------------
// Kernel 1: GRU gate step (fused x-GEMM + h-GEMM + gates).
//   grid = H/16 = 32 blocks; block computes all 128 batch rows of 16 hidden cols.
// ---------------------------------------------------------------------------
__global__ __launch_bounds__(256)
void gru_gates_kernel(const float* __restrict__ xt,     // B x E
                      const float* __restrict__ hprev,  // B x H
                      const float* __restrict__ w_ih,   // 3H x E
                      const float* __restrict__ w_hh,   // 3H x H
                      const float* __restrict__ b_ih,   // 3H
                      const float* __restrict__ b_hh,   // 3H
                      float* __restrict__ hnew)         // B x H
{
    __shared__ float sA[2 * 128 * STR];
    __shared__ float sB[2 * 3 * 16 * STR];

    const int n0   = blockIdx.x * 16;
    const int lane = threadIdx.x & 31;
    const int wave = threadIdx.x >> 5;
    const int l    = lane & 15;
    const int half = lane >> 4;

    // x contribution: accX = {gx_r, gx_z, gx_n}
    v8f accX[3] = { {}, {}, {} };
    block_gemm_lds<3>(xt, EDIM, EDIM,
                      w_ih + (size_t)n0 * EDIM,
                      w_ih + (size_t)(HDIM + n0) * EDIM,
                      w_ih + (size_t)(2 * HDIM + n0) * EDIM,
                      EDIM, accX, sA, sB);

    // h contribution: accH = {gx_r+gh_r, gx_z+gh_z, gh_n}
    v8f accH[3] = { accX[0], accX[1], {} };
    block_gemm_lds<3>(hprev, HDIM, HDIM,
                      w_hh + (size_t)n0 * HDIM,
                      w_hh + (size_t)(HDIM + n0) * HDIM,
                      w_hh + (size_t)(2 * HDIM + n0) * HDIM,
                      HDIM, accH, sA, sB);

    const int   col = n0 + l;
    const float br_ = b_ih[col] + b_hh[col];
    const float bz_ = b_ih[HDIM + col] + b_hh[HDIM + col];
    const float bxn = b_ih[2 * HDIM + col];
    const float bhn = b_hh[2 * HDIM + col];
    #pragma unroll
    for (int v = 0; v < 8; ++v) {
        const int row = wave * 16 + v + 8 * half;   // C/D layout: VGPR v -> M=v (+8 hi half)
        const float r  = sigmoidf_(accH[0][v] + br_);
        const float z  = sigmoidf_(accH[1][v] + bz_);
        const float n  = tanhf(accX[2][v] + bxn + r * (accH[2][v] + bhn));
        const float ho = hprev[row * HDIM + col];
        hnew[row * HDIM + col] = (1.0f - z) * n + z * ho;
    }
}

// ---------------------------------------------------------------------------
// Kernel 2: t = h_new @ w_in.T  (128x512, K=512)
// ---------------------------------------------------------------------------
__global__ __launch_bounds__(256)
void gemm_tvec_kernel(const float* __restrict__ A,   // B x H
                      const float* __restrict__ W,   // H x H row-major
                      float* __restrict__ D)         // B x H
{
    __shared__ float sA[2 * 128 * STR];
    __shared__ float sB[2 * 16 * STR];

    const int n0   = blockIdx.x * 16;
    const int lane = threadIdx.x & 31;
    const int wave = threadIdx.x >> 5;
    const int l    = lane & 15;
    const int half = lane >> 4;

    v8f acc[1] = { {} };
    block_gemm_lds<1>(A, HDIM, HDIM,
                      W + (size_t)n0 * HDIM, nullptr, nullptr,
                      HDIM, acc, sA, sB);

    const int col = n0 + l;
    #pragma unroll
    for (int v = 0; v < 8; ++v) {
        const int row = wave * 16 + v + 8 * half;
        D[row * HDIM + col] = acc[0][v];
    }
}

// ---------------------------------------------------------------------------
// Kernel 3: attention for one step. One block per batch element.
//   Streams ctx (134 MB -> L2-resident) twice, coalesced; shuffle reductions.
// ---------------------------------------------------------------------------
__global__ __launch_bounds__(256)
void attention_kernel(const float* __restrict__ ctx,   // S x B x H
                      const float* __restrict__ tvec,  // B x H
                      float* __restrict__ attn_t,      // B x S (this step's slice)
                      float* __restrict__ wc)          // B x H
{
    __shared__ float tl[HDIM];
    __shared__ float sc[SRC];
    __shared__ float red[256];
    const int b    = blockIdx.x;
    const int tid  = threadIdx.x;
    const int lane = tid & 31;
    const int wave = tid >> 5;

    for (int i = tid; i < HDIM; i += 256) tl[i] = tvec[(size_t)b * HDIM + i];
    __syncthreads();

    for (int s = wave; s < SRC; s += 8) {
        const float* cp = ctx + ((size_t)s * BATCH + b) * HDIM;
        float sum = 0.0f;
        for (int h = lane; h < HDIM; h += 32) sum += cp[h] * tl[h];
        #pragma unroll
        for (int off = 16; off > 0; off >>= 1) sum += __shfl_xor(sum, off, 32);
        if (lane == 0) sc[s] = sum;
    }
    __syncthreads();

    float m = -INFINITY;
    for (int i = tid; i < SRC; i += 256) m = fmaxf(m, sc[i]);
    red[tid] = m;
    __syncthreads();
    for (int off = 128; off > 0; off >>= 1) {
        if (tid < off) red[tid] = fmaxf(red[tid], red[tid + off]);
        __syncthreads();
    }
    const float mx = red[0];
    __syncthreads();

    float psum = 0.0f;
    for (int i = tid; i < SRC; i += 256) {
        const float e = __expf(sc[i] - mx);
        sc[i] = e;
        psum += e;
    }
    red[tid] = psum;
    __syncthreads();
    for (int off = 128; off > 0; off >>= 1) {
        if (tid < off) red[tid] += red[tid + off];
        __syncthreads();
    }
    const float inv = 1.0f / red[0];

    for (int i = tid; i < SRC; i += 256) {
        const float a = sc[i] * inv;
        sc[i] = a;
        attn_t[(size_t)b * SRC + i] = a;
    }
    __syncthreads();

    for (int h0 = tid; h0 < HDIM; h0 += 256) {
        float acc = 0.0f;
        const float* cp = ctx + (size_t)b * HDIM + h0;
        for (int s = 0; s < SRC; ++s) acc += sc[s] * cp[(size_t)s * BATCH * HDIM];
        wc[(size_t)b * HDIM + h0] = acc;
    }
}

// ---------------------------------------------------------------------------
// Kernel 4: out = tanh( [wc | h_new] @ w_out.T )   (128x512, K=1024)
// K split into the two 512-wide halves, same accumulator.
// ---------------------------------------------------------------------------
__global__ __launch_bounds__(256)
void gemm_out_kernel(const float* __restrict__ wc,    // B x H
                     const float* __restrict__ hnew,  // B x H
                     const float* __restrict__ W,     // H x 2H row-major
                     float* __restrict__ out_t)       // B x H
{
    __shared__ float sA[2 * 128 * STR];
    __shared__ float sB[2 * 16 * STR];

    const int n0   = blockIdx.x * 16;
    const int lane = threadIdx.x & 31;
    const int wave = threadIdx.x >> 5;
    const int l    = lane & 15;
    const int half = lane >> 4;

    v8f acc[1] = { {} };
    block_gemm_lds<1>(wc, HDIM, HDIM,
                      W + (size_t)n0 * (2 * HDIM), nullptr, nullptr,
                      2 * HDIM, acc, sA, sB);
    block_gemm_lds<1>(hnew, HDIM, HDIM,
                      W + (size_t)n0 * (2 * HDIM) + HDIM, nullptr, nullptr,
                      2 * HDIM, acc, sA, sB);

    const int col = n0 + l;
    #pragma unroll
    for (int v = 0; v < 8; ++v) {
        const int row = wave * 16 + v + 8 * half;
        out_t[row * HDIM + col] = tanhf(acc[0][v]);
    }
}

// ---------------------------------------------------------------------------
// Kernel 5: p_gen (final step only): sigmoid([wc|h|x] @ w_pgen.T + b)
// ---------------------------------------------------------------------------
__global__ __launch_bounds__(128)
void pgen_kernel(const float* __restrict__ wc,
                 const float* __restrict__ hnew,
                 const float* __restrict__ xlast,   // B x E
                 const float* __restrict__ w_pgen,  // 1 x (2H+E)
                 const float* __restrict__ b_pgen,  // 1
                 float* __restrict__ pg)            // B
{
    __shared__ float red[128];
    const int b = blockIdx.x;
    const int tid = threadIdx.x;
    float s = 0.0f;
    for (int k = tid; k < 2 * HDIM + EDIM; k += 128) {
        float v;
        if (k < HDIM)          v = wc[(size_t)b * HDIM + k];
        else if (k < 2 * HDIM) v = hnew[(size_t)b * HDIM + (k - HDIM)];
        else                   v = xlast[(size_t)b * EDIM + (k - 2 * HDIM)];
        s += w_pgen[k] * v;
    }
    red[tid] = s;
    __syncthreads();
    for (int off = 64; off > 0; off >>= 1) {
        if (tid < off) red[tid] += red[tid + off];
        __syncthreads();
    }
    if (tid == 0) pg[b] = sigmoidf_(red[0] + b_pgen[0]);
}

// ---------------------------------------------------------------------------
extern "C" void kernel_launch(void* const* d_in, const int* in_sizes, int n_in,
                              void* d_out, int out_size, void* d_ws, size_t ws_size,
                              hipStream_t stream) {
    (void)in_sizes; (void)n_in; (void)out_size; (void)ws_size;

    const float* input   = (const float*)d_in[0];   // T x B x E
    const float* hidden  = (const float*)d_in[1];   // 1 x B x H
    const float* context = (const float*)d_in[2];   // S x B x H
    const float* w_ih    = (const float*)d_in[3];   // 3H x E
    const float* w_hh    = (const float*)d_in[4];   // 3H x H
    const float* b_ih    = (const float*)d_in[5];   // 3H
    const float* b_hh    = (const float*)d_in[6];   // 3H
    const float* w_in    = (const float*)d_in[7];   // H x H
    const float* w_out   = (const float*)d_in[8];   // H x 2H
    const float* w_pgen  = (const float*)d_in[9];   // 1 x (2H+E)
    const float* b_pgen  = (const float*)d_in[10];  // 1

    float* outputs = (float*)d_out;                               // T x B x H
    float* attns   = outputs + (size_t)TSTEPS * BATCH * HDIM;     // T x B x S
    float* pg      = attns + (size_t)TSTEPS * BATCH * SRC;        // B

    float* ws = (float*)d_ws;
    float* hbufA = ws;                              // B x H
    float* hbufB = ws + (size_t)BATCH * HDIM;       // B x H
    float* tvec  = ws + (size_t)2 * BATCH * HDIM;   // B x H
    float* wc    = ws + (size_t)3 * BATCH * HDIM;   // B x H
    float* hbuf[2] = { hbufA, hbufB };

    for (int t = 0; t < TSTEPS; ++t) {
        const float* xt = input + (size_t)t * BATCH * EDIM;
        const float* hp = (t == 0) ? hidden : hbuf[(t - 1) & 1];
        float* hn = hbuf[t & 1];

        gru_gates_kernel<<<HDIM / 16, 256, 0, stream>>>(xt, hp, w_ih, w_hh, b_ih, b_hh, hn);
        gemm_tvec_kernel<<<HDIM / 16, 256, 0, stream>>>(hn, w_in, tvec);
        attention_kernel<<<BATCH, 256, 0, stream>>>(context, tvec,
                                                    attns + (size_t)t * BATCH * SRC, wc);
        gemm_out_kernel<<<HDIM / 16, 256, 0, stream>>>(wc, hn, w_out,
                                                       outputs + (size_t)t * BATCH * HDIM);
    }
    pgen_kernel<<<BATCH, 128, 0, stream>>>(wc, hbuf[(TSTEPS - 1) & 1],
                                           input + (size_t)(TSTEPS - 1) * BATCH * EDIM,
                                           w_pgen, b_pgen, pg);
}